// GCNSampling_70669391888552
// MI455X (gfx1250) — compile-verified
//
#include <hip/hip_runtime.h>

typedef __attribute__((ext_vector_type(2))) float v2f;
typedef __attribute__((ext_vector_type(8))) float v8f;

#define NNODES 100000
#define DIN    256
#define HID    16
#define DOUT   64
#define NTILES (NNODES / 16)   // 6250, exact

// ---------------- degree / normalization ----------------

__global__ void k_deg_init(float* __restrict__ deg) {
  int i = blockIdx.x * blockDim.x + threadIdx.x;
  if (i < NNODES) deg[i] = 1.0f;   // self-loop contribution
}

__global__ void k_deg_count(const int* __restrict__ dst, int E,
                            float* __restrict__ deg) {
  int e = blockIdx.x * blockDim.x + threadIdx.x;
  if (e < E) atomicAdd(&deg[dst[e]], 1.0f);
}

__global__ void k_dinv(float* __restrict__ deg) {
  int i = blockIdx.x * blockDim.x + threadIdx.x;
  if (i < NNODES) deg[i] = rsqrtf(deg[i]);   // deg >= 1 always
}

// ---------------- GEMM1: hs1 = dinv[row] * (x @ W1)  [N,256]@[256,16] ----------------

__global__ void __launch_bounds__(256)
k_gemm1(const float* __restrict__ x, const float* __restrict__ W1,
        const float* __restrict__ dinv, float* __restrict__ hs1) {
  __shared__ float lw[DIN * HID];   // 16 KB of the 320 KB WGP LDS
  for (int i = threadIdx.x; i < DIN * HID; i += 256) lw[i] = W1[i];
  __syncthreads();

  const int wave = threadIdx.x >> 5;
  const int lane = threadIdx.x & 31;
  const int half = lane >> 4;
  const int l15  = lane & 15;
  const int tile = blockIdx.x * 8 + wave;
  if (tile >= NTILES) return;       // wave-uniform: EXEC all-ones inside

  const int row = tile * 16 + l15;  // A-matrix row for this lane
  const float* __restrict__ xr = x + (size_t)row * DIN;

  v8f c = {0.f, 0.f, 0.f, 0.f, 0.f, 0.f, 0.f, 0.f};
  #pragma unroll 8
  for (int k0 = 0; k0 < DIN; k0 += 4) {
    const int k = k0 + 2 * half;                 // A: VGPR0<->K=2h, VGPR1<->K=2h+1
    v2f a = *(const v2f*)(xr + k);               // 8B-aligned contiguous load
    v2f b;
    b.x = lw[(k    ) * HID + l15];               // B: K x N, N = lane&15
    b.y = lw[(k + 1) * HID + l15];
    c = __builtin_amdgcn_wmma_f32_16x16x4_f32(false, a, false, b,
                                              (short)0, c, false, false);
  }

  #pragma unroll
  for (int r = 0; r < 8; ++r) {
    const int m = tile * 16 + r + 8 * half;      // C: VGPR r <-> M = r + 8*half
    hs1[(size_t)m * HID + l15] = c[r] * dinv[m];
  }
}

// ---------------- GEMM2: hs2 = dinv[row] * (a1 @ W2)  [N,16]@[16,64] ----------------

__global__ void __launch_bounds__(256)
k_gemm2(const float* __restrict__ a1, const float* __restrict__ W2,
        const float* __restrict__ dinv, float* __restrict__ hs2) {
  __shared__ float lw[HID * DOUT];  // 4 KB
  for (int i = threadIdx.x; i < HID * DOUT; i += 256) lw[i] = W2[i];
  __syncthreads();

  const int wave = threadIdx.x >> 5;
  const int lane = threadIdx.x & 31;
  const int half = lane >> 4;
  const int l15  = lane & 15;
  const int tile = blockIdx.x * 8 + wave;
  if (tile >= NTILES) return;

  const int row = tile * 16 + l15;
  const float* __restrict__ ar = a1 + (size_t)row * HID;

  v8f c0 = {0.f,0.f,0.f,0.f,0.f,0.f,0.f,0.f};
  v8f c1 = c0, c2 = c0, c3 = c0;

  #pragma unroll
  for (int k0 = 0; k0 < HID; k0 += 4) {
    const int k = k0 + 2 * half;
    v2f a = *(const v2f*)(ar + k);               // shared A across 4 N-tiles
    v2f b;
    b.x = lw[(k    ) * DOUT +  0 + l15];
    b.y = lw[(k + 1) * DOUT +  0 + l15];
    c0 = __builtin_amdgcn_wmma_f32_16x16x4_f32(false, a, false, b, (short)0, c0, false, false);
    b.x = lw[(k    ) * DOUT + 16 + l15];
    b.y = lw[(k + 1) * DOUT + 16 + l15];
    c1 = __builtin_amdgcn_wmma_f32_16x16x4_f32(false, a, false, b, (short)0, c1, false, false);
    b.x = lw[(k    ) * DOUT + 32 + l15];
    b.y = lw[(k + 1) * DOUT + 32 + l15];
    c2 = __builtin_amdgcn_wmma_f32_16x16x4_f32(false, a, false, b, (short)0, c2, false, false);
    b.x = lw[(k    ) * DOUT + 48 + l15];
    b.y = lw[(k + 1) * DOUT + 48 + l15];
    c3 = __builtin_amdgcn_wmma_f32_16x16x4_f32(false, a, false, b, (short)0, c3, false, false);
  }

  #pragma unroll
  for (int r = 0; r < 8; ++r) {
    const int m = tile * 16 + r + 8 * half;
    const float s = dinv[m];
    float* __restrict__ o = hs2 + (size_t)m * DOUT;
    o[ 0 + l15] = c0[r] * s;
    o[16 + l15] = c1[r] * s;
    o[32 + l15] = c2[r] * s;
    o[48 + l15] = c3[r] * s;
  }
}

// ---------------- self-loop seed copy ----------------

__global__ void k_copy4(const float4* __restrict__ src, float4* __restrict__ dst, int n4) {
  int i = blockIdx.x * blockDim.x + threadIdx.x;
  if (i < n4) dst[i] = src[i];
}

// ---------------- edge scatter-add (L2-resident atomics) ----------------

// 4 threads per edge, each moves a float4 of the 16 features
__global__ void k_agg1(const int* __restrict__ src, const int* __restrict__ dst, int E,
                       const float* __restrict__ hs1, float* __restrict__ acc1) {
  int t = blockIdx.x * blockDim.x + threadIdx.x;
  int e = t >> 2;
  if (e >= E) return;
  int c = (t & 3) * 4;
  int s = src[e], d = dst[e];
  float4 v = *(const float4*)(hs1 + (size_t)s * HID + c);
  float* __restrict__ q = acc1 + (size_t)d * HID + c;
  atomicAdd(q + 0, v.x);
  atomicAdd(q + 1, v.y);
  atomicAdd(q + 2, v.z);
  atomicAdd(q + 3, v.w);
}

// 16 threads per edge, each moves a float4 of the 64 features
__global__ void k_agg2(const int* __restrict__ src, const int* __restrict__ dst, int E,
                       const float* __restrict__ hs2, float* __restrict__ out) {
  int t = blockIdx.x * blockDim.x + threadIdx.x;
  int e = t >> 4;
  if (e >= E) return;
  int c = (t & 15) * 4;
  int s = src[e], d = dst[e];
  float4 v = *(const float4*)(hs2 + (size_t)s * DOUT + c);
  float* __restrict__ q = out + (size_t)d * DOUT + c;
  atomicAdd(q + 0, v.x);
  atomicAdd(q + 1, v.y);
  atomicAdd(q + 2, v.z);
  atomicAdd(q + 3, v.w);
}

// ---------------- epilogues ----------------

// a1 = relu(dinv[i] * acc1 + b1)  (in place)
__global__ void k_post1(float* __restrict__ acc1, const float* __restrict__ dinv,
                        const float* __restrict__ b1) {
  int t = blockIdx.x * blockDim.x + threadIdx.x;
  if (t >= NNODES * HID) return;
  int i = t >> 4, f = t & 15;
  acc1[t] = fmaxf(dinv[i] * acc1[t] + b1[f], 0.0f);
}

// out = dinv[i] * out + b2  (in place)
__global__ void k_post2(float* __restrict__ out, const float* __restrict__ dinv,
                        const float* __restrict__ b2) {
  int t = blockIdx.x * blockDim.x + threadIdx.x;
  if (t >= NNODES * DOUT) return;
  int i = t >> 6, f = t & 63;
  out[t] = dinv[i] * out[t] + b2[f];
}

// ---------------- launch ----------------

extern "C" void kernel_launch(void* const* d_in, const int* in_sizes, int n_in,
                              void* d_out, int out_size, void* d_ws, size_t ws_size,
                              hipStream_t stream) {
  const float* x  = (const float*)d_in[0];
  const int*   ei = (const int*)d_in[1];
  const float* W1 = (const float*)d_in[2];
  const float* b1 = (const float*)d_in[3];
  const float* W2 = (const float*)d_in[4];
  const float* b2 = (const float*)d_in[5];
  const int E = in_sizes[1] / 2;
  const int* srcp = ei;
  const int* dstp = ei + E;

  float* ws   = (float*)d_ws;
  float* dinv = ws;                              // N floats
  float* hs1  = dinv + NNODES;                   // N*16
  float* acc1 = hs1 + (size_t)NNODES * HID;      // N*16
  float* hs2  = acc1 + (size_t)NNODES * HID;     // N*64
  float* out  = (float*)d_out;                   // N*64

  // normalization: dinv = rsqrt(in_degree + 1)
  k_deg_init <<<(NNODES + 255) / 256, 256, 0, stream>>>(dinv);
  k_deg_count<<<(E + 255) / 256,      256, 0, stream>>>(dstp, E, dinv);
  k_dinv     <<<(NNODES + 255) / 256, 256, 0, stream>>>(dinv);

  // layer 1
  k_gemm1<<<(NTILES + 7) / 8, 256, 0, stream>>>(x, W1, dinv, hs1);
  k_copy4<<<(NNODES * HID / 4 + 255) / 256, 256, 0, stream>>>(
      (const float4*)hs1, (float4*)acc1, NNODES * HID / 4);
  k_agg1 <<<((E * 4) + 255) / 256, 256, 0, stream>>>(srcp, dstp, E, hs1, acc1);
  k_post1<<<(NNODES * HID + 255) / 256, 256, 0, stream>>>(acc1, dinv, b1);

  // layer 2
  k_gemm2<<<(NTILES + 7) / 8, 256, 0, stream>>>(acc1, W2, dinv, hs2);
  k_copy4<<<(NNODES * DOUT / 4 + 255) / 256, 256, 0, stream>>>(
      (const float4*)hs2, (float4*)out, NNODES * DOUT / 4);
  k_agg2 <<<((E * 16) + 255) / 256, 256, 0, stream>>>(srcp, dstp, E, hs2, out);
  k_post2<<<(NNODES * DOUT + 255) / 256, 256, 0, stream>>>(out, dinv, b2);
}